// Model_59528246723051
// MI455X (gfx1250) — compile-verified
//
#include <hip/hip_runtime.h>
#include <math.h>

// ---------------------------------------------------------------------------
// PCN-style autoencoder forward on MI455X (gfx1250), fp32 WMMA 16x16x4.
// B=16, N_IN=N_OUT=2048, LATENT=256, INTER=1024. P = B*N = 32768.
// Register-blocked WMMA GEMMs + async-to-LDS staging in Chamfer.
// ---------------------------------------------------------------------------

#define BATCH   16
#define NPTS    2048
#define PTOT    32768          // BATCH*NPTS, = 2^15
#define LATENT  256
#define INTER   1024
#define FC3OUT  6144

typedef __attribute__((ext_vector_type(2))) float v2f;
typedef __attribute__((ext_vector_type(8))) float v8f;
typedef __attribute__((ext_vector_type(4))) int   v4i;

#define AS1 __attribute__((address_space(1)))
#define AS3 __attribute__((address_space(3)))

__device__ __forceinline__ v8f wmma_f32_k4(v2f a, v2f b, v8f c) {
  // D = A(16x4,f32) * B(4x16,f32) + C(16x16,f32); wave32, EXEC all ones.
  return __builtin_amdgcn_wmma_f32_16x16x4_f32(
      /*neg_a=*/false, a, /*neg_b=*/false, b,
      /*c_mod=*/(short)0, c, /*reuse_a=*/false, /*reuse_b=*/false);
}

#if __has_builtin(__builtin_amdgcn_global_load_async_to_lds_b128)
#define USE_ASYNC_LDS 1
#endif

// ---------------- normalization -------------------------------------------
__global__ void norm_stats(const float* __restrict__ x,
                           float* __restrict__ com, float* __restrict__ scale) {
  int b = blockIdx.x;
  const float* xb = x + (size_t)b * NPTS * 3;
  float sx = 0.f, sy = 0.f, sz = 0.f;
  for (int i = threadIdx.x; i < NPTS; i += 256) {
    sx += xb[i * 3 + 0]; sy += xb[i * 3 + 1]; sz += xb[i * 3 + 2];
  }
  __shared__ float sh[3][256];
  sh[0][threadIdx.x] = sx; sh[1][threadIdx.x] = sy; sh[2][threadIdx.x] = sz;
  __syncthreads();
  for (int off = 128; off > 0; off >>= 1) {
    if (threadIdx.x < off) {
      sh[0][threadIdx.x] += sh[0][threadIdx.x + off];
      sh[1][threadIdx.x] += sh[1][threadIdx.x + off];
      sh[2][threadIdx.x] += sh[2][threadIdx.x + off];
    }
    __syncthreads();
  }
  __shared__ float cmx, cmy, cmz;
  if (threadIdx.x == 0) {
    cmx = sh[0][0] / NPTS; cmy = sh[1][0] / NPTS; cmz = sh[2][0] / NPTS;
  }
  __syncthreads();
  float mx = 0.f;
  for (int i = threadIdx.x; i < NPTS; i += 256) {
    float dx = xb[i * 3 + 0] - cmx, dy = xb[i * 3 + 1] - cmy, dz = xb[i * 3 + 2] - cmz;
    mx = fmaxf(mx, sqrtf(dx * dx + dy * dy + dz * dz));
  }
  __shared__ float shm[256];
  shm[threadIdx.x] = mx; __syncthreads();
  for (int off = 128; off > 0; off >>= 1) {
    if (threadIdx.x < off) shm[threadIdx.x] = fmaxf(shm[threadIdx.x], shm[threadIdx.x + off]);
    __syncthreads();
  }
  if (threadIdx.x == 0) {
    com[b * 3 + 0] = cmx; com[b * 3 + 1] = cmy; com[b * 3 + 2] = cmz;
    scale[b] = shm[0];
  }
}

// X0[c][p] channel-major (stride PTOT), channel 3 zero-padded for K=4 WMMA.
__global__ void build_x0(const float* __restrict__ x, const float* __restrict__ com,
                         const float* __restrict__ scale, float* __restrict__ X0) {
  int p = blockIdx.x * 256 + threadIdx.x;      // p = b*NPTS + n
  int b = p >> 11;
  float inv = 1.0f / scale[b];
#pragma unroll
  for (int d = 0; d < 3; ++d)
    X0[(size_t)d * PTOT + p] = (x[(size_t)p * 3 + d] - com[b * 3 + d]) * inv;
  X0[(size_t)3 * PTOT + p] = 0.f;
}

__global__ void pad_w1(const float* __restrict__ w1, float* __restrict__ W1p) {
  int idx = threadIdx.x;                       // 64*4 = 256 entries
  int r = idx >> 2, c = idx & 3;
  W1p[idx] = (c < 3) ? w1[r * 3 + c] : 0.f;
}

__global__ void zero_buf(float* __restrict__ p, int n) {
  int i = blockIdx.x * 256 + threadIdx.x;
  if (i < n) p[i] = 0.f;
}

// ---------------- encoder GEMM: H[Cout][P] = W[Cout][Cin] * Xin[Cin][P] ----
// Register blocked: each wave computes 4 M-tiles x 2 N-tiles (64x32 output),
// reusing B-frags across M-tiles and A-frags across N-tiles (1.5 dword/WMMA).
__global__ void enc_gemm(const float* __restrict__ W, const float* __restrict__ Xin,
                         float* __restrict__ Hout, int Cin, int Cout) {
  int lane = threadIdx.x & 31;
  int wave = threadIdx.x >> 5;
  int p0 = blockIdx.x * 256 + wave * 32;       // 2 N-tiles: p0, p0+16
  int m0 = blockIdx.y * 64;                    // 4 M-tiles
  int half = lane >> 4;                        // 0: lanes 0-15, 1: lanes 16-31
  int l = lane & 15;
  int klo = half * 2;                          // A/B frag K offset per ISA layout
  v8f acc[4][2] = {};
  for (int k0 = 0; k0 < Cin; k0 += 4) {
    const float* xr = Xin + (size_t)(k0 + klo) * PTOT + p0 + l;
    v2f b0, b1;
    b0.x = xr[0];  b0.y = xr[PTOT];
    b1.x = xr[16]; b1.y = xr[PTOT + 16];
#pragma unroll
    for (int t = 0; t < 4; ++t) {
      const float* wr = W + (size_t)(m0 + t * 16 + l) * Cin + k0 + klo;
      v2f a; a.x = wr[0]; a.y = wr[1];
      acc[t][0] = wmma_f32_k4(a, b0, acc[t][0]);
      acc[t][1] = wmma_f32_k4(a, b1, acc[t][1]);
    }
  }
#pragma unroll
  for (int t = 0; t < 4; ++t) {
#pragma unroll
    for (int j = 0; j < 8; ++j) {              // C/D: VGPR j -> M = j + half*8
      size_t row = (size_t)(m0 + t * 16 + j + half * 8) * PTOT;
      Hout[row + p0 + l]      = acc[t][0][j];
      Hout[row + p0 + 16 + l] = acc[t][1][j];
    }
  }
  (void)Cout;
}

// ---------------- BatchNorm (training mode, over B*N samples/channel) ------
__global__ void bn_stats(const float* __restrict__ H, float* __restrict__ sums,
                         float* __restrict__ sumsq) {
  int c = blockIdx.y;
  const float* row = H + (size_t)c * PTOT + blockIdx.x * 8192;
  float s = 0.f, sq = 0.f;
  for (int i = threadIdx.x; i < 8192; i += 256) {
    float v = row[i];
    s += v; sq += v * v;
  }
  __shared__ float sh[2][256];
  sh[0][threadIdx.x] = s; sh[1][threadIdx.x] = sq;
  __syncthreads();
  for (int off = 128; off > 0; off >>= 1) {
    if (threadIdx.x < off) {
      sh[0][threadIdx.x] += sh[0][threadIdx.x + off];
      sh[1][threadIdx.x] += sh[1][threadIdx.x + off];
    }
    __syncthreads();
  }
  if (threadIdx.x == 0) {
    atomicAdd(&sums[c], sh[0][0]);
    atomicAdd(&sumsq[c], sh[1][0]);
  }
}

__global__ void bn_finalize(const float* __restrict__ sums, const float* __restrict__ sumsq,
                            const float* __restrict__ g, const float* __restrict__ be,
                            float* __restrict__ ab, int Cout) {
  int c = threadIdx.x;
  if (c >= Cout) return;
  const float invP = 1.0f / (float)PTOT;
  float mean = sums[c] * invP;
  float var = sumsq[c] * invP - mean * mean;
  float a = g[c] * rsqrtf(var + 1e-5f);
  ab[c] = a;
  ab[Cout + c] = be[c] - mean * a;             // encoder conv-bias cancels in train-mode BN
}

__global__ void bn_apply_relu(float* __restrict__ H, const float* __restrict__ ab, int Cout) {
  size_t idx = (size_t)blockIdx.x * 256 + threadIdx.x;
  int c = (int)(idx >> 15);                    // / PTOT
  float v = fmaf(ab[c], H[idx], ab[Cout + c]);
  H[idx] = v > 0.f ? v : 0.f;
}

// ---------------- global max pool: z[b][c] = max_n H[c][b*N+n] -------------
__global__ void maxpool(const float* __restrict__ H, float* __restrict__ z) {
  int c = blockIdx.x, b = blockIdx.y;
  const float* row = H + (size_t)c * PTOT + b * NPTS;
  float m = -3.4e38f;
  for (int i = threadIdx.x; i < NPTS; i += 256) m = fmaxf(m, row[i]);
  __shared__ float sh[256];
  sh[threadIdx.x] = m; __syncthreads();
  for (int off = 128; off > 0; off >>= 1) {
    if (threadIdx.x < off) sh[threadIdx.x] = fmaxf(sh[threadIdx.x], sh[threadIdx.x + off]);
    __syncthreads();
  }
  if (threadIdx.x == 0) z[b * LATENT + c] = sh[0];
}

// ---------------- decoder FC: out[16][O] = act[16][K] @ W[O][K]^T + bias ---
// One wave per 4 output tiles (64 outputs); A-frag reused across the 4 WMMAs.
__global__ void fc_gemm(const float* __restrict__ act, const float* __restrict__ W,
                        const float* __restrict__ bias, float* __restrict__ out,
                        int K, int O, int relu) {
  int lane = threadIdx.x & 31;
  int wave = threadIdx.x >> 5;
  int grp = (blockIdx.x * 8 + wave) * 64;      // 4 tiles of 16 outputs
  if (grp >= O) return;                        // uniform per-wave; O % 512 == 0 here
  int half = lane >> 4, l = lane & 15, klo = half * 2;
  v8f acc[4] = {};
  for (int k0 = 0; k0 < K; k0 += 4) {
    const float* ar = act + (size_t)l * K + k0 + klo;      // A[m=l][k]
    v2f a; a.x = ar[0]; a.y = ar[1];
#pragma unroll
    for (int t = 0; t < 4; ++t) {
      const float* wr = W + (size_t)(grp + t * 16 + l) * K + k0 + klo; // B[k][n]=W[n][k]
      v2f b; b.x = wr[0]; b.y = wr[1];
      acc[t] = wmma_f32_k4(a, b, acc[t]);
    }
  }
#pragma unroll
  for (int t = 0; t < 4; ++t) {
#pragma unroll
    for (int j = 0; j < 8; ++j) {
      int m = j + half * 8;
      int n = grp + t * 16 + l;
      float v = acc[t][j] + bias[n];
      if (relu) v = v > 0.f ? v : 0.f;
      out[(size_t)m * O + n] = v;
    }
  }
}

// pred[b][n][d] = h3[b][d*2048+n] * scale[b] + com[b][d]
__global__ void make_pred(const float* __restrict__ h3, const float* __restrict__ com,
                          const float* __restrict__ scale, float* __restrict__ pred) {
  int idx = blockIdx.x * 256 + threadIdx.x;    // b*NPTS + n
  int b = idx >> 11, n = idx & 2047;
  float s = scale[b];
#pragma unroll
  for (int d = 0; d < 3; ++d)
    pred[(size_t)idx * 3 + d] = h3[(size_t)b * FC3OUT + d * NPTS + n] * s + com[b * 3 + d];
}

// ---------------- Chamfer: per query in A, min sq-dist to set Bp -----------
// Ref-point tiles staged to LDS via GLOBAL_LOAD_ASYNC_TO_LDS_B128 (ASYNCcnt)
// when available; next tile prefetched with global_prefetch_b8.
__global__ void chamfer_min(const float* __restrict__ A, const float* __restrict__ Bp,
                            float* __restrict__ dmin) {
  int b = blockIdx.y;
  int q = blockIdx.x * 128 + threadIdx.x;
  const float* Ab = A + (size_t)b * NPTS * 3;
  const float* Bb = Bp + (size_t)b * NPTS * 3;
  float qx = Ab[q * 3 + 0], qy = Ab[q * 3 + 1], qz = Ab[q * 3 + 2];
  __shared__ __align__(16) float tile[128 * 3];
  float best = 3.4e38f;
  for (int t0 = 0; t0 < NPTS; t0 += 128) {
#ifdef USE_ASYNC_LDS
    if (threadIdx.x < 96) {                    // 96 lanes x 16B = 384 floats
      const float* gsrc = Bb + t0 * 3 + threadIdx.x * 4;
      float* ldst = &tile[threadIdx.x * 4];
      __builtin_amdgcn_global_load_async_to_lds_b128(
          (AS1 v4i*)(unsigned long long)gsrc,
          (AS3 v4i*)(unsigned long long)ldst,
          0, 0);
    }
#if __has_builtin(__builtin_amdgcn_s_wait_asynccnt)
    __builtin_amdgcn_s_wait_asynccnt(0);
#else
    asm volatile("s_wait_asynccnt 0x0" ::: "memory");
#endif
#else
    for (int i = threadIdx.x; i < 384; i += 128) tile[i] = Bb[t0 * 3 + i];
#endif
    __syncthreads();
    if (t0 + 128 < NPTS)                       // -> global_prefetch_b8
      __builtin_prefetch(&Bb[(t0 + 128) * 3 + threadIdx.x * 3], 0, 1);
#pragma unroll 4
    for (int j = 0; j < 128; ++j) {
      float dx = qx - tile[j * 3 + 0];
      float dy = qy - tile[j * 3 + 1];
      float dz = qz - tile[j * 3 + 2];
      best = fminf(best, dx * dx + dy * dy + dz * dz);
    }
    __syncthreads();
  }
  dmin[b * NPTS + q] = best;
}

__global__ void cd_reduce(const float* __restrict__ d1, const float* __restrict__ d2,
                          float* __restrict__ cd_l1, float* __restrict__ cd_l2) {
  int b = blockIdx.x;
  float s1 = 0.f, s2 = 0.f, s3 = 0.f, s4 = 0.f;
  for (int i = threadIdx.x; i < NPTS; i += 256) {
    float a = d1[b * NPTS + i], c = d2[b * NPTS + i];
    s1 += sqrtf(a); s2 += a;
    s3 += sqrtf(c); s4 += c;
  }
  __shared__ float sh[4][256];
  sh[0][threadIdx.x] = s1; sh[1][threadIdx.x] = s2;
  sh[2][threadIdx.x] = s3; sh[3][threadIdx.x] = s4;
  __syncthreads();
  for (int off = 128; off > 0; off >>= 1) {
    if (threadIdx.x < off)
#pragma unroll
      for (int k = 0; k < 4; ++k) sh[k][threadIdx.x] += sh[k][threadIdx.x + off];
    __syncthreads();
  }
  if (threadIdx.x == 0) {
    const float invN = 1.0f / (float)NPTS;
    cd_l1[b] = (sh[0][0] * invN + sh[2][0] * invN) * 0.5f;
    cd_l2[b] = sh[1][0] * invN + sh[3][0] * invN;
  }
}

__global__ void loss_kernel(const float* __restrict__ cd_l2, float* __restrict__ loss) {
  if (threadIdx.x == 0) {
    float s = 0.f;
    for (int i = 0; i < BATCH; ++i) s += cd_l2[i];
    *loss = s / (float)BATCH;
  }
}

// ---------------------------------------------------------------------------
extern "C" void kernel_launch(void* const* d_in, const int* in_sizes, int n_in,
                              void* d_out, int out_size, void* d_ws, size_t ws_size,
                              hipStream_t stream) {
  const float* x  = (const float*)d_in[0];
  const float* gt = (const float*)d_in[1];
  // d_in[2] = labels (unused by reference forward)

  const float *w[5], *g[5], *be[5];
  const float *fc1w, *fc1b, *fc2w, *fc2b, *fc3w, *fc3b;
  if (n_in >= 29 && in_sizes[3] == 192) {
    // insertion order: w1,b1,g1,be1, w2,..., fc1_w,fc1_b, fc2_w,fc2_b, fc3_w,fc3_b
    for (int i = 0; i < 5; ++i) {
      w[i]  = (const float*)d_in[3 + 4 * i];
      g[i]  = (const float*)d_in[5 + 4 * i];
      be[i] = (const float*)d_in[6 + 4 * i];
    }
    fc1w = (const float*)d_in[23]; fc1b = (const float*)d_in[24];
    fc2w = (const float*)d_in[25]; fc2b = (const float*)d_in[26];
    fc3w = (const float*)d_in[27]; fc3b = (const float*)d_in[28];
  } else {
    // jax tree_flatten alphabetical: b1..b5, be1..be5, fc1_b,fc1_w,fc2_b,fc2_w,fc3_b,fc3_w, g1..g5, w1..w5
    for (int i = 0; i < 5; ++i) {
      be[i] = (const float*)d_in[8 + i];
      g[i]  = (const float*)d_in[19 + i];
      w[i]  = (const float*)d_in[24 + i];
    }
    fc1b = (const float*)d_in[13]; fc1w = (const float*)d_in[14];
    fc2b = (const float*)d_in[15]; fc2w = (const float*)d_in[16];
    fc3b = (const float*)d_in[17]; fc3w = (const float*)d_in[18];
  }

  // ---- workspace layout (floats) ----
  float* ws = (float*)d_ws;
  float* X0    = ws;                                  // 4*PTOT
  float* bufA  = X0 + (size_t)4 * PTOT;               // 256*PTOT
  float* bufB  = bufA + (size_t)256 * PTOT;           // 256*PTOT
  float* small = bufB + (size_t)256 * PTOT;
  float* sums  = small;                               // 5*256
  float* sumsq = sums + 5 * 256;                      // 5*256
  float* ab    = sumsq + 5 * 256;                     // 5*512
  float* W1p   = ab + 5 * 512;                        // 256
  float* com   = W1p + 256;                           // 48
  float* scal  = com + 48;                            // 16
  float* z     = scal + 16;                           // 16*256
  float* h1    = z + BATCH * LATENT;                  // 16*1024
  float* h2    = h1 + BATCH * INTER;                  // 16*2048
  float* h3    = h2 + BATCH * NPTS;                   // 16*6144
  float* dst1  = h3 + BATCH * FC3OUT;                 // 16*2048
  float* dst2  = dst1 + BATCH * NPTS;                 // 16*2048

  float* pred  = (float*)d_out;                       // 98304
  float* lossp = pred + BATCH * NPTS * 3;             // 1
  float* cdl1  = lossp + 1;                           // 16
  float* cdl2  = cdl1 + BATCH;                        // 16

  // ---- preprocessing ----
  zero_buf<<<10, 256, 0, stream>>>(sums, 2 * 5 * 256);
  pad_w1<<<1, 256, 0, stream>>>(w[0], W1p);
  norm_stats<<<BATCH, 256, 0, stream>>>(x, com, scal);
  build_x0<<<PTOT / 256, 256, 0, stream>>>(x, com, scal, X0);

  // ---- encoder: 5 x (WMMA GEMM -> BN stats -> finalize -> apply+ReLU) ----
  const int cin_[5]  = {4, 64, 128, 128, 256};        // layer1 K padded to 4
  const int cout_[5] = {64, 128, 128, 256, 256};
  const float* xin = X0;
  float* hbuf[5] = {bufA, bufB, bufA, bufB, bufA};
  for (int i = 0; i < 5; ++i) {
    const float* Wi = (i == 0) ? W1p : w[i];
    float* Hout = hbuf[i];
    enc_gemm<<<dim3(PTOT / 256, cout_[i] / 64), 256, 0, stream>>>(Wi, xin, Hout, cin_[i], cout_[i]);
    bn_stats<<<dim3(PTOT / 8192, cout_[i]), 256, 0, stream>>>(Hout, sums + i * 256, sumsq + i * 256);
    bn_finalize<<<1, 256, 0, stream>>>(sums + i * 256, sumsq + i * 256, g[i], be[i],
                                       ab + i * 512, cout_[i]);
    bn_apply_relu<<<cout_[i] * (PTOT / 256), 256, 0, stream>>>(Hout, ab + i * 512, cout_[i]);
    xin = Hout;
  }

  // ---- max pool -> z[16][256] ----
  maxpool<<<dim3(LATENT, BATCH), 256, 0, stream>>>(hbuf[4], z);

  // ---- decoder (WMMA, M=16 single tile row; 4 O-tiles per wave) ----
  fc_gemm<<<(INTER / 64) / 8, 256, 0, stream>>>(z,  fc1w, fc1b, h1, LATENT, INTER, 1);
  fc_gemm<<<(NPTS  / 64) / 8, 256, 0, stream>>>(h1, fc2w, fc2b, h2, INTER,  NPTS,  1);
  fc_gemm<<<(FC3OUT/ 64) / 8, 256, 0, stream>>>(h2, fc3w, fc3b, h3, NPTS,   FC3OUT, 0);

  // ---- pred + Chamfer + reductions ----
  make_pred<<<(BATCH * NPTS) / 256, 256, 0, stream>>>(h3, com, scal, pred);
  chamfer_min<<<dim3(NPTS / 128, BATCH), 128, 0, stream>>>(pred, gt, dst1);
  chamfer_min<<<dim3(NPTS / 128, BATCH), 128, 0, stream>>>(gt, pred, dst2);
  cd_reduce<<<BATCH, 256, 0, stream>>>(dst1, dst2, cdl1, cdl2);
  loss_kernel<<<1, 32, 0, stream>>>(cdl2, lossp);

  (void)out_size; (void)ws_size;
}